// MultiHeadLinear_72825465471143
// MI455X (gfx1250) — compile-verified
//
#include <hip/hip_runtime.h>
#include <hip/hip_bf16.h>

// MultiHeadLinear for MI455X (gfx1250):
//   y[node, :] = x[node, :] @ kernel[head[graph(node)]] + bias[head[graph(node)]]
// 256 graphs x 64 nodes, IN=OUT=128, f32 end-to-end via V_WMMA_F32_16X16X4_F32.
// Tile staging uses CDNA5 GLOBAL_LOAD_ASYNC_TO_LDS_B128 (ASYNCcnt path).

#define N_GRAPHS        256
#define NODES_PER_GRAPH 64
#define IN_F            128
#define OUT_F           128

typedef __attribute__((ext_vector_type(2))) float v2f;
typedef __attribute__((ext_vector_type(8))) float v8f;

// LDS tiles with bank-conflict padding (64 banks x 4B on CDNA5):
//  Xs row stride 132 floats -> A-frag b64 reads cover 64 distinct banks
//  Ws row stride 136 floats -> the two lane-half row reads use disjoint bank sets
#define XS_STRIDE 132
#define WS_STRIDE 136

__global__ __launch_bounds__(256)
void mhl_wmma_f32_kernel(const float* __restrict__ inputs,   // [16384, 128]
                         const int*   __restrict__ head,     // [256]
                         const float* __restrict__ kernel_w, // [16, 128, 128]
                         const float* __restrict__ bias,     // [16, 128]
                         float*       __restrict__ out)      // [16384, 128]
{
    __shared__ float Xs[NODES_PER_GRAPH][XS_STRIDE]; // 64 x 132 x 4B = 33,792 B
    __shared__ float Ws[IN_F][WS_STRIDE];            // 128 x 136 x 4B = 69,632 B

    const int g    = blockIdx.x;          // graph id
    const int tid  = threadIdx.x;         // 0..255
    const int wave = tid >> 5;            // 0..7 (wave32)
    const int lane = tid & 31;
    const int l16  = lane & 15;
    const int lh   = lane >> 4;           // lane half: 0 or 1

    const int h = head[g];
    const float* __restrict__ X = inputs   + (size_t)g * NODES_PER_GRAPH * IN_F;
    const float* __restrict__ W = kernel_w + (size_t)h * IN_F * OUT_F;
    const float* __restrict__ B = bias     + (size_t)h * OUT_F;

    // ---- Stage X tile (64x128) into LDS via async DMA: 2048 float4, 8/thread ----
    {
        #pragma unroll
        for (int i = 0; i < 8; ++i) {
            int idx4 = i * 256 + tid;               // float4 index within tile
            int row  = idx4 >> 5;                   // 32 float4 per row
            int c4   = (idx4 & 31) << 2;            // starting float column
            unsigned ldsoff = (unsigned)(size_t)&Xs[row][c4];   // wave-relative LDS byte addr
            unsigned goff   = (unsigned)(idx4 << 4);            // global byte offset
            asm volatile("global_load_async_to_lds_b128 %0, %1, %2"
                         :: "v"(ldsoff), "v"(goff), "s"(X) : "memory");
        }
    }
    // ---- Stage W (128x128) into LDS via async DMA: 4096 float4, 16/thread ----
    {
        #pragma unroll
        for (int i = 0; i < 16; ++i) {
            int idx4 = i * 256 + tid;
            int row  = idx4 >> 5;
            int c4   = (idx4 & 31) << 2;
            unsigned ldsoff = (unsigned)(size_t)&Ws[row][c4];
            unsigned goff   = (unsigned)(idx4 << 4);
            asm volatile("global_load_async_to_lds_b128 %0, %1, %2"
                         :: "v"(ldsoff), "v"(goff), "s"(W) : "memory");
        }
    }
    // Drain this wave's async copies, then rendezvous with the other 7 waves.
    asm volatile("s_wait_asynccnt 0x0" ::: "memory");
    __syncthreads();

    // ---- Tile assignment: wave w -> row block (w&3), col blocks (w>>2)*4 .. +3 ----
    const int wr  = wave & 3;             // output row block (16 rows each)
    const int wc  = (wave >> 2) << 2;     // first of 4 column blocks
    const int am  = wr * 16 + l16;        // A-fragment row (node within graph)

    v8f acc0 = {0.f,0.f,0.f,0.f,0.f,0.f,0.f,0.f};
    v8f acc1 = acc0, acc2 = acc0, acc3 = acc0;

    const int n0 = (wc + 0) * 16 + l16;   // per-lane output columns
    const int n1 = (wc + 1) * 16 + l16;
    const int n2 = (wc + 2) * 16 + l16;
    const int n3 = (wc + 3) * 16 + l16;

    // ---- K loop: 128 = 32 steps of K=4; A frag reused across 4 WMMAs ----
    #pragma unroll 8
    for (int kk = 0; kk < IN_F / 4; ++kk) {
        const int k0 = kk * 4 + lh * 2;   // this lane-half's K pair

        // A 16x4 f32 frag: a[v] = X[am][k0+v] -> one b64 LDS read
        v2f a = *(const v2f*)&Xs[am][k0];

        // B 4x16 f32 frags: b[v] = W[k0+v][n]
        v2f b0; b0.x = Ws[k0][n0]; b0.y = Ws[k0 + 1][n0];
        v2f b1; b1.x = Ws[k0][n1]; b1.y = Ws[k0 + 1][n1];
        v2f b2; b2.x = Ws[k0][n2]; b2.y = Ws[k0 + 1][n2];
        v2f b3; b3.x = Ws[k0][n3]; b3.y = Ws[k0 + 1][n3];

        acc0 = __builtin_amdgcn_wmma_f32_16x16x4_f32(false, a, false, b0, (short)0, acc0, false, false);
        acc1 = __builtin_amdgcn_wmma_f32_16x16x4_f32(false, a, false, b1, (short)0, acc1, false, false);
        acc2 = __builtin_amdgcn_wmma_f32_16x16x4_f32(false, a, false, b2, (short)0, acc2, false, false);
        acc3 = __builtin_amdgcn_wmma_f32_16x16x4_f32(false, a, false, b3, (short)0, acc3, false, false);
    }

    // ---- Epilogue: add bias, store. d[r] = D[r + 8*lh][l16] ----
    const float bv0 = B[n0];
    const float bv1 = B[n1];
    const float bv2 = B[n2];
    const float bv3 = B[n3];

    float* __restrict__ Y = out + (size_t)g * NODES_PER_GRAPH * OUT_F;

    #pragma unroll
    for (int r = 0; r < 8; ++r) {
        const int m   = r + 8 * lh;               // row within 16x16 tile
        const int row = wr * 16 + m;              // row within graph tile
        float* yrow = Y + (size_t)row * OUT_F;
        yrow[n0] = acc0[r] + bv0;
        yrow[n1] = acc1[r] + bv1;
        yrow[n2] = acc2[r] + bv2;
        yrow[n3] = acc3[r] + bv3;
    }
}

extern "C" void kernel_launch(void* const* d_in, const int* in_sizes, int n_in,
                              void* d_out, int out_size, void* d_ws, size_t ws_size,
                              hipStream_t stream) {
    const float* inputs   = (const float*)d_in[0]; // [16384,128] f32
    // d_in[1] = n_node [256] int32 (fixed at 64 per graph; unused)
    const int*   head     = (const int*)  d_in[2]; // [256] int32
    const float* kernel_w = (const float*)d_in[3]; // [16,128,128] f32
    const float* bias     = (const float*)d_in[4]; // [16,128] f32
    float*       out      = (float*)d_out;         // [16384,128] f32

    (void)in_sizes; (void)n_in; (void)out_size; (void)d_ws; (void)ws_size;

    mhl_wmma_f32_kernel<<<dim3(N_GRAPHS), dim3(256), 0, stream>>>(
        inputs, head, kernel_w, bias, out);
}